// GNN_45217415693100
// MI455X (gfx1250) — compile-verified
//
#include <hip/hip_runtime.h>

// GCN sizes (fixed by the reference)
#define NN  100000
#define EE  1600000
#define IND 128
#define HD  128
#define OD  64

typedef __attribute__((ext_vector_type(2))) float v2f;
typedef __attribute__((ext_vector_type(8))) float v8f;

// ---------------- degree / normalization ----------------

__global__ void k_init_deg(float* __restrict__ deg) {
  int i = blockIdx.x * blockDim.x + threadIdx.x;
  if (i < NN) deg[i] = 1.0f;  // self-loop counts once
}

__global__ void k_deg_accum(const int* __restrict__ dst, float* __restrict__ deg) {
  int e = blockIdx.x * blockDim.x + threadIdx.x;
  if (e < EE) unsafeAtomicAdd(&deg[dst[e]], 1.0f);
}

__global__ void k_dinv(float* __restrict__ deg) {
  int i = blockIdx.x * blockDim.x + threadIdx.x;
  if (i < NN) deg[i] = rsqrtf(deg[i]);  // deg >= 1 always (self-loops)
}

// ---------------- GEMM1: h1 = x @ W1 (fp32 WMMA) ----------------
// One wave owns a 16x16 tile: wave w -> columns [16w,16w+16), 8 waves cover H=128.
// Grid-stride over 16-row strips; B fragments (W1 column tile) hoisted & reused.

__global__ __launch_bounds__(256) void k_gemm1(const float* __restrict__ x,
                                               const float* __restrict__ W,
                                               float* __restrict__ h) {
  const int wave = threadIdx.x >> 5;
  const int lane = threadIdx.x & 31;
  const int lh   = lane >> 4;   // lane half: selects K pair (ISA 7.12.2, 32-bit A 16x4)
  const int l15  = lane & 15;
  const int n0   = wave << 4;

  v2f bf[32];
#pragma unroll
  for (int kk = 0; kk < 32; ++kk) {
    const int k = (kk << 2) + (lh << 1);
    bf[kk].x = W[(k + 0) * HD + n0 + l15];
    bf[kk].y = W[(k + 1) * HD + n0 + l15];
  }

  for (int m0 = blockIdx.x << 4; m0 < NN; m0 += (int)(gridDim.x << 4)) {
    const float* xr = x + (size_t)(m0 + l15) * IND;
    v8f c = {};
#pragma unroll
    for (int kk = 0; kk < 32; ++kk) {
      const int k = (kk << 2) + (lh << 1);
      v2f a;
      a.x = xr[k];
      a.y = xr[k + 1];
      c = __builtin_amdgcn_wmma_f32_16x16x4_f32(false, a, false, bf[kk],
                                                (short)0, c, false, false);
    }
#pragma unroll
    for (int r = 0; r < 8; ++r)
      h[(size_t)(m0 + r + (lh << 3)) * HD + n0 + l15] = c[r];
  }
}

// ---------------- GEMM2: h2 = relu(agg1 + b1) @ W2 (fp32 WMMA) ----------------
// 8 waves: (wave>>2) picks 16-row substrip, (wave&3) picks one of 4 column tiles (OD=64).
// bias+ReLU fused into the A-fragment load (pure function of loaded value -> safe).

__global__ __launch_bounds__(256) void k_gemm2(const float* __restrict__ agg,
                                               const float* __restrict__ b1,
                                               const float* __restrict__ W,
                                               float* __restrict__ h2) {
  const int wave = threadIdx.x >> 5;
  const int lane = threadIdx.x & 31;
  const int lh   = lane >> 4;
  const int l15  = lane & 15;
  const int n0   = (wave & 3) << 4;
  const int mo   = (wave >> 2) << 4;

  v2f bf[32];
#pragma unroll
  for (int kk = 0; kk < 32; ++kk) {
    const int k = (kk << 2) + (lh << 1);
    bf[kk].x = W[(k + 0) * OD + n0 + l15];
    bf[kk].y = W[(k + 1) * OD + n0 + l15];
  }

  for (int m0 = (blockIdx.x << 5) + mo; m0 < NN; m0 += (int)(gridDim.x << 5)) {
    const float* ar = agg + (size_t)(m0 + l15) * HD;
    v8f c = {};
#pragma unroll
    for (int kk = 0; kk < 32; ++kk) {
      const int k = (kk << 2) + (lh << 1);
      v2f a;
      a.x = fmaxf(ar[k]     + b1[k],     0.0f);
      a.y = fmaxf(ar[k + 1] + b1[k + 1], 0.0f);
      c = __builtin_amdgcn_wmma_f32_16x16x4_f32(false, a, false, bf[kk],
                                                (short)0, c, false, false);
    }
#pragma unroll
    for (int r = 0; r < 8; ++r)
      h2[(size_t)(m0 + r + (lh << 3)) * OD + n0 + l15] = c[r];
  }
}

// ---------------- self-loop inits (also zero-initialize accumulators) ----------------

__global__ void k_selfloop_h(const float* __restrict__ h, const float* __restrict__ dinv,
                             float* __restrict__ agg) {
  int t = blockIdx.x * blockDim.x + threadIdx.x;     // one thread per float4
  const int per = HD / 4;
  int i = t / per, c4 = (t % per) << 2;
  if (i < NN) {
    float s = dinv[i]; s *= s;                       // norm of self-loop = 1/deg
    const float4 v = *(const float4*)(h + (size_t)i * HD + c4);
    float4 o; o.x = v.x * s; o.y = v.y * s; o.z = v.z * s; o.w = v.w * s;
    *(float4*)(agg + (size_t)i * HD + c4) = o;
  }
}

__global__ void k_selfloop_out(const float* __restrict__ h2, const float* __restrict__ dinv,
                               const float* __restrict__ b2, float* __restrict__ out) {
  int t = blockIdx.x * blockDim.x + threadIdx.x;
  const int per = OD / 4;
  int i = t / per, c4 = (t % per) << 2;
  if (i < NN) {
    float s = dinv[i]; s *= s;
    const float4 v = *(const float4*)(h2 + (size_t)i * OD + c4);
    float4 o;
    o.x = b2[c4 + 0] + v.x * s;
    o.y = b2[c4 + 1] + v.y * s;
    o.z = b2[c4 + 2] + v.z * s;
    o.w = b2[c4 + 3] + v.w * s;
    *(float4*)(out + (size_t)i * OD + c4) = o;
  }
}

// ---------------- edge scatter (L2-resident gather + fp32 atomics) ----------------
// Layer 1: 32 lanes per edge, each lane handles 4 features (float4 gather, 4 atomics).

__global__ void k_scatter1(const int* __restrict__ src, const int* __restrict__ dst,
                           const float* __restrict__ dinv, const float* __restrict__ h,
                           float* __restrict__ agg) {
  int t = blockIdx.x * blockDim.x + threadIdx.x;
  int e = t >> 5, ln = t & 31;
  if (e >= EE) return;
  int s = src[e], d = dst[e];
  float nrm = dinv[s] * dinv[d];
  const float4 v = *(const float4*)(h + (size_t)s * HD + (ln << 2));
  float* p = agg + (size_t)d * HD + (ln << 2);
  unsafeAtomicAdd(p + 0, v.x * nrm);
  unsafeAtomicAdd(p + 1, v.y * nrm);
  unsafeAtomicAdd(p + 2, v.z * nrm);
  unsafeAtomicAdd(p + 3, v.w * nrm);
}

// Layer 2: 16 lanes per edge (OD=64).
__global__ void k_scatter2(const int* __restrict__ src, const int* __restrict__ dst,
                           const float* __restrict__ dinv, const float* __restrict__ h2,
                           float* __restrict__ out) {
  int t = blockIdx.x * blockDim.x + threadIdx.x;
  int e = t >> 4, ln = t & 15;
  if (e >= EE) return;
  int s = src[e], d = dst[e];
  float nrm = dinv[s] * dinv[d];
  const float4 v = *(const float4*)(h2 + (size_t)s * OD + (ln << 2));
  float* p = out + (size_t)d * OD + (ln << 2);
  unsafeAtomicAdd(p + 0, v.x * nrm);
  unsafeAtomicAdd(p + 1, v.y * nrm);
  unsafeAtomicAdd(p + 2, v.z * nrm);
  unsafeAtomicAdd(p + 3, v.w * nrm);
}

// ---------------- host launch ----------------

extern "C" void kernel_launch(void* const* d_in, const int* in_sizes, int n_in,
                              void* d_out, int out_size, void* d_ws, size_t ws_size,
                              hipStream_t stream) {
  const float* x  = (const float*)d_in[0];
  const int*   ei = (const int*)d_in[1];    // (2, E): [0]=src, [1]=dst
  const float* W1 = (const float*)d_in[2];
  const float* b1 = (const float*)d_in[3];
  const float* W2 = (const float*)d_in[4];
  const float* b2 = (const float*)d_in[5];
  float* out = (float*)d_out;

  const int* src = ei;
  const int* dst = ei + EE;

  char* ws = (char*)d_ws;
  size_t off = 0;
  auto carve = [&](size_t bytes) -> float* {
    float* p = (float*)(ws + off);
    off = (off + bytes + 4095) & ~(size_t)4095;
    return p;
  };
  float* dinv = carve((size_t)NN * 4);            // deg -> dinv (in place)
  float* h1   = carve((size_t)NN * HD * 4);       // x @ W1
  float* agg1 = carve((size_t)NN * HD * 4);       // normalized aggregation of h1
  float* h2   = carve((size_t)NN * OD * 4);       // relu(agg1+b1) @ W2
  (void)ws_size; (void)n_in; (void)in_sizes; (void)out_size;

  const int T = 256;

  // 1) degrees (self-loop baked in), then deg^{-1/2}
  k_init_deg <<<(NN + T - 1) / T, T, 0, stream>>>(dinv);
  k_deg_accum<<<(EE + T - 1) / T, T, 0, stream>>>(dst, dinv);
  k_dinv     <<<(NN + T - 1) / T, T, 0, stream>>>(dinv);

  // 2) h1 = x @ W1  (fp32 WMMA, grid-stride over 16-row strips)
  k_gemm1<<<2048, 256, 0, stream>>>(x, W1, h1);

  // 3) agg1 = selfloop(h1) then += edge messages (fp32 atomics, L2-resident)
  k_selfloop_h<<<(NN * (HD / 4) + T - 1) / T, T, 0, stream>>>(h1, dinv, agg1);
  k_scatter1  <<<(EE * 32) / T, T, 0, stream>>>(src, dst, dinv, h1, agg1);

  // 4) h2 = relu(agg1 + b1) @ W2 (bias+ReLU fused into A-fragment load)
  k_gemm2<<<2048, 256, 0, stream>>>(agg1, b1, W2, h2);

  // 5) out = b2 + selfloop(h2) then += edge messages
  k_selfloop_out<<<(NN * (OD / 4) + T - 1) / T, T, 0, stream>>>(h2, dinv, b2, out);
  k_scatter2    <<<(EE * 16) / T, T, 0, stream>>>(src, dst, dinv, h2, out);
}